// PointNetSetAbstraction_6133213299364
// MI455X (gfx1250) — compile-verified
//
#include <hip/hip_runtime.h>
#include <hip/hip_bf16.h>

// PointNet SetAbstraction for MI455X (gfx1250, wave32).
// GEMMs run on the matrix pipe via V_WMMA_F32_16X16X4_F32 (full f32 — the op
// is bandwidth bound, so no precision is sacrificed for speed we don't need).
// BN(batch-stats) is a global reduction; normalize+ReLU is folded into the
// next layer's operand load. All reductions are fixed-order -> deterministic.
// kNN is one-wave-per-query: lane-private LDS slices + shfl_xor butterfly
// argmin -> zero workgroup barriers in the selection loop.
//
// Workspace requirement: ~407 MB
//   [0,2MB)        knn indices  [B,S,K] int32
//   [2MB,2MB+64KB) per-(channel,chunk) partial sums
//   [+2KB]         per-channel affine (A,B) for the 3 layers
//   [4MB, +134MB)  bufA : X0 (gathered input), later reused as Y1
//   [.., +268MB)   bufB : Y0 (layer0 out), later reused as Y2 (layer2 out)

typedef __attribute__((ext_vector_type(2))) float v2f;
typedef __attribute__((ext_vector_type(8))) float v8f;

constexpr int    BB = 16, NN = 4096, SS = 1024, KK = 32, DD = 64;
constexpr size_t MM = (size_t)BB * SS * KK;   // 524288 GEMM columns
constexpr float  BN_EPS = 1e-5f;
constexpr float  INFF   = 3.402823466e+38f;

// ---------------------------------------------------------------------------
// 1) Gather new_xyz / new_normals and copy fps_idx to output.
// ---------------------------------------------------------------------------
__global__ void gather_out_kernel(const float* __restrict__ xyz,
                                  const float* __restrict__ normals,
                                  const int*   __restrict__ fps_idx,
                                  float* __restrict__ o_xyz,
                                  float* __restrict__ o_norm,
                                  int*   __restrict__ o_idx) {
  int t = blockIdx.x * blockDim.x + threadIdx.x;
  if (t >= BB * SS) return;
  int b = t / SS;
  int n = fps_idx[t];
  size_t src = ((size_t)b * NN + (size_t)n) * 3;
  size_t dst = (size_t)t * 3;
  o_xyz[dst + 0]  = xyz[src + 0];
  o_xyz[dst + 1]  = xyz[src + 1];
  o_xyz[dst + 2]  = xyz[src + 2];
  o_norm[dst + 0] = normals[src + 0];
  o_norm[dst + 1] = normals[src + 1];
  o_norm[dst + 2] = normals[src + 2];
  o_idx[t] = n;
}

// ---------------------------------------------------------------------------
// 2) kNN: one wave32 per query, 4 queries per 128-thread WG.
//    Lane L owns candidates n = L + 32*j (lane-private LDS slice -> no
//    barriers). 32 argmin rounds via shfl_xor butterfly on (val,idx) pairs;
//    tie-break on lower index matches jax.lax.top_k ordering. Per-lane
//    candidates cached as 4 chunk minima so a removal rescans only 32 entries.
// ---------------------------------------------------------------------------
__global__ __launch_bounds__(128)
void knn_kernel(const float* __restrict__ xyz,
                const int*   __restrict__ fps_idx,
                int*         __restrict__ idx_out) {
  extern __shared__ float lds_dist[];           // 4 waves * 4096 floats = 64 KB

  int tid  = threadIdx.x;
  int wave = tid >> 5;
  int lane = tid & 31;

  int bs = blockIdx.x * 4 + wave;               // b*S + s
  int b  = bs / SS;

  float* dist = lds_dist + (size_t)wave * NN;
  const float* xb = xyz + (size_t)b * NN * 3;

  int   qi = fps_idx[bs];
  float qx = xb[qi * 3 + 0], qy = xb[qi * 3 + 1], qz = xb[qi * 3 + 2];

  // |p|^2 - 2 q.p  (dropping per-query constant |q|^2; argmin unchanged).
  // Lane-private stores: dist[lane + 32*j].
  for (int j = 0; j < NN / 32; ++j) {
    int n = lane + 32 * j;
    float px = xb[n * 3 + 0], py = xb[n * 3 + 1], pz = xb[n * 3 + 2];
    dist[n] = px * px + py * py + pz * pz - 2.0f * (qx * px + qy * py + qz * pz);
  }

  // 4 chunk minima per lane, chunk cc covers j in [cc*32, cc*32+32)
  float cmin[4]; int cidx[4];
  #pragma unroll
  for (int cc = 0; cc < 4; ++cc) {
    float lm = INFF; int li = NN;
    for (int j = cc * 32; j < cc * 32 + 32; ++j) {
      int n = lane + 32 * j;
      float v = dist[n];
      if (v < lm) { lm = v; li = n; }           // ascending scan keeps lowest idx on tie
    }
    cmin[cc] = lm; cidx[cc] = li;
  }

  for (int kk = 0; kk < KK; ++kk) {
    // lane min over the 4 cached chunks (lexicographic, tie -> lower idx)
    float v = cmin[0]; int ix = cidx[0];
    #pragma unroll
    for (int cc = 1; cc < 4; ++cc) {
      if (cmin[cc] < v || (cmin[cc] == v && cidx[cc] < ix)) { v = cmin[cc]; ix = cidx[cc]; }
    }
    // wave32 butterfly argmin: all lanes converge on the same (v, ix)
    #pragma unroll
    for (int off = 16; off > 0; off >>= 1) {
      float vo = __shfl_xor(v,  off, 32);
      int   io = __shfl_xor(ix, off, 32);
      if (vo < v || (vo == v && io < ix)) { v = vo; ix = io; }
    }
    int sel = ix;                               // uniform across wave
    if (lane == 0) idx_out[(size_t)bs * KK + kk] = sel;

    // owner lane invalidates + rescans only the affected 32-entry chunk
    if ((sel & 31) == lane) {
      dist[sel] = INFF;
      int cc = (sel >> 5) >> 5;                 // j = sel>>5 ; chunk = j>>5
      float lm = INFF; int li = NN;
      for (int j = cc * 32; j < cc * 32 + 32; ++j) {
        int n = lane + 32 * j;
        float dv = dist[n];
        if (dv < lm) { lm = dv; li = n; }
      }
      cmin[cc] = lm; cidx[cc] = li;
    }
  }
}

// ---------------------------------------------------------------------------
// 3) Gather grouped points into GEMM layout X0[c][m], m=((b*S+s)*K+kk).
//    Writes are coalesced across m; reads are L2-resident gathers.
// ---------------------------------------------------------------------------
__global__ void gather_points_kernel(const float* __restrict__ points,
                                     const int*   __restrict__ idx_knn,
                                     float*       __restrict__ X0) {
  size_t m = (size_t)blockIdx.x * blockDim.x + threadIdx.x;
  if (m >= MM) return;
  int b = (int)(m / ((size_t)SS * KK));
  int n = idx_knn[m];
  const float* src = points + ((size_t)b * NN + (size_t)n) * DD;
  #pragma unroll 8
  for (int c = 0; c < DD; ++c) X0[(size_t)c * MM + m] = src[c];
}

// ---------------------------------------------------------------------------
// 4) 1x1 conv as WMMA f32 GEMM:  Y[C_out, M] = W[C_out,64] * X[64, M] + bias.
//    RT = C_out/16 row tiles. 8 waves/WG, each wave owns one 16-col stripe.
//    AFFINE folds previous layer's BN+ReLU into the operand load.
//    A-frag (16x4 f32): lane<16 -> rows, K={k,k+1}; lane>=16 -> K={k+2,k+3}.
//    B-frag (4x16 f32): mirrored K split across lane halves.
// ---------------------------------------------------------------------------
template <int RT, int AFFINE>
__global__ __launch_bounds__(256)
void conv_wmma_kernel(const float* __restrict__ Xin,
                      const float* __restrict__ W,       // [RT*16][64] row-major
                      const float* __restrict__ bconv,   // [RT*16]
                      const float* __restrict__ Aff,     // [64] (per in-channel)
                      const float* __restrict__ Bff,     // [64]
                      float* __restrict__ Yout) {
  __shared__ float Wl[RT * 16 * 64];
  __shared__ float Al[64];
  __shared__ float Bl[64];
  __shared__ float biasl[RT * 16];

  int tid = threadIdx.x;
  // float4 staging: b128 loads + ds_store_b128
  for (int i = tid * 4; i < RT * 16 * 64; i += 256 * 4) {
    *(float4*)&Wl[i] = *(const float4*)&W[i];
  }
  if (AFFINE) { if (tid < 64) { Al[tid] = Aff[tid]; Bl[tid] = Bff[tid]; } }
  if (tid < RT * 16) biasl[tid] = bconv[tid];
  __syncthreads();

  int wave = tid >> 5, lane = tid & 31;
  int lh = lane & 15, sel = lane >> 4;
  size_t col = ((size_t)blockIdx.x * 8 + wave) * 16 + lh;

  v8f acc[RT] = {};

  #pragma unroll
  for (int k = 0; k < 64; k += 4) {
    int rowb = k + 2 * sel;                       // K-pair this lane half holds
    float bx = Xin[(size_t)rowb * MM + col];
    float by = Xin[(size_t)(rowb + 1) * MM + col];
    if (AFFINE) {                                 // fused BN + ReLU of prev layer
      bx = fmaxf(fmaf(Al[rowb],     bx, Bl[rowb]),     0.0f);
      by = fmaxf(fmaf(Al[rowb + 1], by, Bl[rowb + 1]), 0.0f);
    }
    v2f bf; bf.x = bx; bf.y = by;
    #pragma unroll
    for (int rt = 0; rt < RT; ++rt) {
      int row = rt * 16 + lh;
      v2f af; af.x = Wl[row * 64 + rowb]; af.y = Wl[row * 64 + rowb + 1];
      acc[rt] = __builtin_amdgcn_wmma_f32_16x16x4_f32(
          /*neg_a=*/false, af, /*neg_b=*/false, bf,
          /*c_mod=*/(short)0, acc[rt], /*reuse_a=*/false, /*reuse_b=*/false);
    }
  }

  // C/D 16x16 f32: vgpr r -> row r (lanes 0-15) / row r+8 (lanes 16-31)
  #pragma unroll
  for (int rt = 0; rt < RT; ++rt) {
    #pragma unroll
    for (int r = 0; r < 8; ++r) {
      int row = rt * 16 + r + 8 * sel;
      Yout[(size_t)row * MM + col] = acc[rt][r] + biasl[row];
    }
  }
}

// ---------------------------------------------------------------------------
// 5) BN batch stats: per-channel sum/sumsq over M, chunked (deterministic,
//    no float atomics), then finalize into fused affine A=rstd*g,
//    B=bt-mean*A for the next layer / maxpool.
// ---------------------------------------------------------------------------
__global__ __launch_bounds__(256)
void stats_partial_kernel(const float* __restrict__ Y,
                          float* __restrict__ partial) {   // [C][64][2]
  __shared__ float ssum[256], ssum2[256];
  int chunk = blockIdx.x;   // 0..63
  int c     = blockIdx.y;
  int tid   = threadIdx.x;
  const int CHUNK = (int)(MM / 64);   // 8192
  size_t base = (size_t)c * MM + (size_t)chunk * CHUNK;
  float s = 0.0f, s2 = 0.0f;
  for (int i = tid; i < CHUNK; i += 256) {
    float v = Y[base + i];
    s += v; s2 += v * v;
  }
  ssum[tid] = s; ssum2[tid] = s2;
  __syncthreads();
  for (int off = 128; off > 0; off >>= 1) {
    if (tid < off) { ssum[tid] += ssum[tid + off]; ssum2[tid] += ssum2[tid + off]; }
    __syncthreads();
  }
  if (tid == 0) {
    size_t o = ((size_t)c * 64 + chunk) * 2;
    partial[o] = ssum[0]; partial[o + 1] = ssum2[0];
  }
}

__global__ void stats_final_kernel(const float* __restrict__ partial,
                                   const float* __restrict__ g,
                                   const float* __restrict__ bt,
                                   float* __restrict__ A,
                                   float* __restrict__ Bc, int C) {
  int c = blockIdx.x * blockDim.x + threadIdx.x;
  if (c >= C) return;
  float s = 0.0f, s2 = 0.0f;
  for (int ch = 0; ch < 64; ++ch) {           // fixed order -> deterministic
    s  += partial[((size_t)c * 64 + ch) * 2];
    s2 += partial[((size_t)c * 64 + ch) * 2 + 1];
  }
  float inv  = 1.0f / (float)MM;
  float mean = s * inv;
  float var  = fmaxf(s2 * inv - mean * mean, 0.0f);   // biased var (jnp.var)
  float rstd = rsqrtf(var + BN_EPS);
  float a = rstd * g[c];
  A[c]  = a;
  Bc[c] = bt[c] - mean * a;
}

// ---------------------------------------------------------------------------
// 6) Max over K with fused BN+ReLU of layer 2; feat[b,s,c] row-major.
//    Each thread owns a contiguous 128 B run -> b128 loads.
// ---------------------------------------------------------------------------
__global__ __launch_bounds__(128)
void maxpool_kernel(const float* __restrict__ Y2,
                    const float* __restrict__ A,
                    const float* __restrict__ Bc,
                    float* __restrict__ feat) {
  int c = threadIdx.x;                        // 0..127
  size_t m0 = (size_t)blockIdx.x * KK;        // (b*S+s)*K, contiguous K run
  const float4* p = (const float4*)(Y2 + (size_t)c * MM + m0);
  float a = A[c], bb = Bc[c];
  float mx = -INFF;
  #pragma unroll
  for (int q = 0; q < KK / 4; ++q) {
    float4 v = p[q];
    mx = fmaxf(mx, fmaxf(fmaf(a, v.x, bb), 0.0f));
    mx = fmaxf(mx, fmaxf(fmaf(a, v.y, bb), 0.0f));
    mx = fmaxf(mx, fmaxf(fmaf(a, v.z, bb), 0.0f));
    mx = fmaxf(mx, fmaxf(fmaf(a, v.w, bb), 0.0f));
  }
  feat[(size_t)blockIdx.x * 128 + c] = mx;
}

// ---------------------------------------------------------------------------
extern "C" void kernel_launch(void* const* d_in, const int* in_sizes, int n_in,
                              void* d_out, int out_size, void* d_ws, size_t ws_size,
                              hipStream_t stream) {
  (void)in_sizes; (void)n_in; (void)out_size; (void)ws_size;

  const float* xyz     = (const float*)d_in[0];
  const float* normals = (const float*)d_in[1];
  const float* points  = (const float*)d_in[2];
  const int*   fps_idx = (const int*)  d_in[3];
  const float* w0 = (const float*)d_in[4],  *b0 = (const float*)d_in[5];
  const float* g0 = (const float*)d_in[6],  *bt0 = (const float*)d_in[7];
  const float* w1 = (const float*)d_in[8],  *b1 = (const float*)d_in[9];
  const float* g1 = (const float*)d_in[10], *bt1 = (const float*)d_in[11];
  const float* w2 = (const float*)d_in[12], *b2 = (const float*)d_in[13];
  const float* g2 = (const float*)d_in[14], *bt2 = (const float*)d_in[15];

  // Output layout: new_xyz [B,S,3] | new_normals [B,S,3] | feat [B,S,128] | fps_idx [B,S]
  float* out    = (float*)d_out;
  float* o_xyz  = out;
  float* o_norm = out + (size_t)BB * SS * 3;
  float* o_feat = out + (size_t)BB * SS * 6;
  int*   o_idx  = (int*)(out + (size_t)BB * SS * 6 + (size_t)BB * SS * 128);

  // Workspace layout
  char*  ws      = (char*)d_ws;
  int*   idx_knn = (int*)ws;                                   // 2 MB
  float* partial = (float*)(ws + (2u << 20));                  // 64 KB
  float* AB      = (float*)(ws + (2u << 20) + (64u << 10));    // 2 KB
  float *A0 = AB,       *B0 = AB + 64;
  float *A1 = AB + 128, *B1 = AB + 192;
  float *A2 = AB + 256, *B2 = AB + 384;
  float* bufA = (float*)(ws + (4u << 20));                         // 134 MB: X0 -> Y1
  float* bufB = (float*)(ws + (4u << 20) + (size_t)64 * MM * 4);   // 268 MB: Y0 -> Y2

  // 1) direct output gathers
  gather_out_kernel<<<(BB * SS + 255) / 256, 256, 0, stream>>>(
      xyz, normals, fps_idx, o_xyz, o_norm, o_idx);

  // 2) kNN indices: 4 queries per WG (one wave each), 64 KB dynamic LDS
  knn_kernel<<<(BB * SS) / 4, 128, 4 * NN * sizeof(float), stream>>>(
      xyz, fps_idx, idx_knn);

  // 3) group features into GEMM layout
  gather_points_kernel<<<(int)((MM + 255) / 256), 256, 0, stream>>>(points, idx_knn, bufA);

  const int GEMM_BLOCKS = (int)(MM / 128);   // 8 waves * 16 cols per WG

  // layer 0: X0(bufA) -> Y0(bufB), no input affine
  conv_wmma_kernel<4, 0><<<GEMM_BLOCKS, 256, 0, stream>>>(bufA, w0, b0, nullptr, nullptr, bufB);
  stats_partial_kernel<<<dim3(64, 64), 256, 0, stream>>>(bufB, partial);
  stats_final_kernel<<<1, 128, 0, stream>>>(partial, g0, bt0, A0, B0, 64);

  // layer 1: Y0(bufB) -> Y1(bufA), fused BN0+ReLU on input
  conv_wmma_kernel<4, 1><<<GEMM_BLOCKS, 256, 0, stream>>>(bufB, w1, b1, A0, B0, bufA);
  stats_partial_kernel<<<dim3(64, 64), 256, 0, stream>>>(bufA, partial);
  stats_final_kernel<<<1, 128, 0, stream>>>(partial, g1, bt1, A1, B1, 64);

  // layer 2: Y1(bufA) -> Y2(bufB), fused BN1+ReLU on input, C_out=128
  conv_wmma_kernel<8, 1><<<GEMM_BLOCKS, 256, 0, stream>>>(bufA, w2, b2, A1, B1, bufB);
  stats_partial_kernel<<<dim3(64, 128), 256, 0, stream>>>(bufB, partial);
  stats_final_kernel<<<1, 128, 0, stream>>>(partial, g2, bt2, A2, B2, 128);

  // 4) fused BN2+ReLU + max over K
  maxpool_kernel<<<BB * SS, 128, 0, stream>>>(bufB, A2, B2, o_feat);
}